// TDTPrefixSearch_67310727463187
// MI455X (gfx1250) — compile-verified
//
#include <hip/hip_runtime.h>
#include <hip/hip_bf16.h>
#include <utility>

#define WAVE 32
#define NDUR 5
#define NEGBIG (-1.0e30f)

// ---------------- CDNA5 async global->LDS path (guarded) ----------------
#if defined(__HIP_DEVICE_COMPILE__) && __has_builtin(__builtin_amdgcn_global_load_async_to_lds_b32)
#define USE_ASYNC_LDS 1
typedef __attribute__((address_space(1))) int glob_int;   // "__device__ int *"
typedef __attribute__((address_space(3))) int lds_int;    // "__shared__ int *"

// Issue one 4 KB tile (32 lanes x b32 x 32 instrs) with constant IOFFSETs.
template <int... J>
__device__ __forceinline__ void issue_tile_impl(const float* g, float* l,
                                                std::integer_sequence<int, J...>) {
  (__builtin_amdgcn_global_load_async_to_lds_b32(
       (glob_int*)g, (lds_int*)l, J * 128, 0), ...);
}
__device__ __forceinline__ void issue_tile(const float* g, float* l) {
  issue_tile_impl(g, l, std::make_integer_sequence<int, 32>{});
}
#else
#define USE_ASYNC_LDS 0
#endif

__device__ __forceinline__ void wait_async0() {
#if defined(__HIP_DEVICE_COMPILE__)
#if __has_builtin(__builtin_amdgcn_s_wait_asynccnt)
  __builtin_amdgcn_s_wait_asynccnt(0);
#else
  asm volatile("s_wait_asynccnt 0" ::: "memory");
#endif
#endif
}

// =======================================================================
// Phase 1: per-(t,u) logsumexp over V, then emit/blank log-probs.
// One wave32 per cell; 4 waves/block; grid-stride, 8 cells per wave.
// =======================================================================
__global__ __launch_bounds__(128) void tdt_lse_kernel(
    const float* __restrict__ logits, const int* __restrict__ targets,
    float* __restrict__ emit_ws, float* __restrict__ blank_ws,
    int U, int D, int V, int npairs)
{
  const int lane   = threadIdx.x & (WAVE - 1);
  const int wave   = threadIdx.x >> 5;
  const int stride = gridDim.x * 4;
  int p = blockIdx.x * 4 + wave;

#if USE_ASYNC_LDS
  __shared__ float lbuf[4][2][1024];   // 32 KB: per-wave double buffer
  if (V == 1024) {
    int buf = 0;
    if (p < npairs)
      issue_tile(logits + (size_t)p * D + lane, &lbuf[wave][buf][lane]);
    for (; p < npairs; p += stride) {
      wait_async0();                       // current tile resident in LDS
      asm volatile("" ::: "memory");
      int pn = p + stride;                 // kick off next tile DMA now;
      if (pn < npairs)                     // it overlaps the reduction below
        issue_tile(logits + (size_t)pn * D + lane, &lbuf[wave][buf ^ 1][lane]);
      const float* tile = lbuf[wave][buf];
      float m = NEGBIG;
      #pragma unroll 8
      for (int j = 0; j < 1024; j += 32) m = fmaxf(m, tile[lane + j]);
      #pragma unroll
      for (int off = 16; off; off >>= 1) m = fmaxf(m, __shfl_xor(m, off));
      float s = 0.f;
      #pragma unroll 8
      for (int j = 0; j < 1024; j += 32) s += __expf(tile[lane + j] - m);
      #pragma unroll
      for (int off = 16; off; off >>= 1) s += __shfl_xor(s, off);
      float lse = m + __logf(s);
      if (lane == 0) {
        int u = p % U;
        int tg = targets[u];
        emit_ws[p]  = tile[tg] - lse;      // gathers hit LDS, not HBM
        blank_ws[p] = tile[V - 1] - lse;
      }
      buf ^= 1;
    }
    return;
  }
#endif
  // Fallback: direct global two-pass reduction (4B-aligned b32 streams).
  for (; p < npairs; p += stride) {
    const float* base = logits + (size_t)p * D;
    if (p + stride < npairs)   // -> global_prefetch_b8, one line per lane
      __builtin_prefetch(logits + (size_t)(p + stride) * D + lane * 32, 0, 0);
    float m = NEGBIG;
    for (int j = lane; j < V; j += WAVE) m = fmaxf(m, base[j]);
    #pragma unroll
    for (int off = 16; off; off >>= 1) m = fmaxf(m, __shfl_xor(m, off));
    float s = 0.f;
    for (int j = lane; j < V; j += WAVE) s += __expf(base[j] - m);
    #pragma unroll
    for (int off = 16; off; off >>= 1) s += __shfl_xor(s, off);
    float lse = m + __logf(s);
    if (lane == 0) {
      int u = p % U;
      emit_ws[p]  = base[targets[u]] - lse;
      blank_ws[p] = base[V - 1] - lse;
    }
  }
}

// =======================================================================
// Phase 2: sequential DP over T rows; lane = u. Row recurrence solved as
// max-plus scan: a_u = E_u + prefmax_k(c_k - E_k) with argmax bookkeeping.
// =======================================================================
__global__ __launch_bounds__(WAVE) void tdt_dp_kernel(
    const float* __restrict__ emit_ws, const float* __restrict__ blank_ws,
    const int* __restrict__ tlens, float* __restrict__ out, int T, int U)
{
  const int lane = threadIdx.x;
  const int ulen = tlens[0];
  float* la_out = out + 1;
  int*   st_out = ((int*)out) + 1 + T;
  int*   to_out = ((int*)out) + 1 + 2 * T;

  float pa = 0.f, ps = 0.f, pn = 0.f, blp = 0.f;   // prev row alpha/start/total, prev blank

  auto step = [&](int t, float bl_cur, float e) {
    // inclusive prefix sum of e (e==0 at lane 0)
    float E = e;
    #pragma unroll
    for (int off = 1; off < WAVE; off <<= 1) {
      float v = __shfl_up(E, off);
      if (lane >= off) E += v;
    }
    // d_k = c_k - E_k ; boundary u==0 contributes d=0 ; t==0 forces chain from boundary
    float d;
    if (t == 0) d = (lane == 0) ? 0.f : NEGBIG;
    else        d = (lane == 0) ? 0.f : (pa + blp - E);
    // prefix max + argmax; keep-current-on-tie == prefer larger k (matches >=)
    float bd = d; int bk = lane;
    #pragma unroll
    for (int off = 1; off < WAVE; off <<= 1) {
      float od = __shfl_up(bd, off);
      int   ok = __shfl_up(bk, off);
      if (lane >= off && od > bd) { bd = od; bk = ok; }
    }
    float a  = E + bd;
    float gs = __shfl(ps, bk);
    float gn = __shfl(pn, bk);
    float s, n;
    if (bk == 0) { s = (float)t; n = (float)(lane + 1); }           // chain from u==0 boundary
    else         { s = gs;       n = gn + 1.0f + (float)(lane - bk); }
    if (lane == ulen) {
      float la = a + bl_cur;
      la_out[t] = la;
      st_out[t] = (int)s;
      to_out[t] = (int)(n + 1.0f);
      if (t == T - 1) out[0] = la;
    }
    pa = a; ps = s; pn = n; blp = bl_cur;
  };

  const int CH = 8;
  float emA[CH], blA[CH], emB[CH], blB[CH];
  const bool lane_ok = (lane < U);
  const bool em_ok   = (lane > 0) && lane_ok;
  const int nCh = T / CH;

  #pragma unroll
  for (int k = 0; k < CH; ++k) {                       // preload chunk 0
    blA[k] = lane_ok ? blank_ws[(size_t)k * U + lane]    : 0.f;
    emA[k] = em_ok   ? emit_ws[(size_t)k * U + lane - 1] : 0.f;
  }
  for (int c = 0; c < nCh; ++c) {
    if (c + 2 < nCh) {                                 // L2->near prefetch 2 chunks ahead
      __builtin_prefetch(&blank_ws[(size_t)(c + 2) * CH * U + lane], 0, 1);
      __builtin_prefetch(&emit_ws [(size_t)(c + 2) * CH * U + lane], 0, 1);
    }
    if (c + 1 < nCh) {                                 // register double-buffer next chunk
      int tb = (c + 1) * CH;
      #pragma unroll
      for (int k = 0; k < CH; ++k) {
        blB[k] = lane_ok ? blank_ws[(size_t)(tb + k) * U + lane]    : 0.f;
        emB[k] = em_ok   ? emit_ws[(size_t)(tb + k) * U + lane - 1] : 0.f;
      }
    }
    #pragma unroll
    for (int k = 0; k < CH; ++k) step(c * CH + k, blA[k], emA[k]);
    #pragma unroll
    for (int k = 0; k < CH; ++k) { blA[k] = blB[k]; emA[k] = emB[k]; }
  }
  for (int t = nCh * CH; t < T; ++t) {                 // tail (T % CH != 0)
    float bl = lane_ok ? blank_ws[(size_t)t * U + lane]    : 0.f;
    float e  = em_ok   ? emit_ws[(size_t)t * U + lane - 1] : 0.f;
    step(t, bl, e);
  }
}

extern "C" void kernel_launch(void* const* d_in, const int* in_sizes, int n_in,
                              void* d_out, int out_size, void* d_ws, size_t ws_size,
                              hipStream_t stream) {
  const float* logits  = (const float*)d_in[0];
  const int*   targets = (const int*)d_in[1];
  const int*   tlens   = (const int*)d_in[3];   // target_lens
  (void)n_in; (void)out_size; (void)ws_size;

  const int T = in_sizes[4];                    // t2skip: (T,)
  const int U = in_sizes[1];                    // targets: (1, U)
  const int D = (int)((long long)in_sizes[0] / ((long long)T * U));
  const int V = D - NDUR;
  const int npairs = T * U;

  float* emit_ws  = (float*)d_ws;               // [T*U]
  float* blank_ws = emit_ws + npairs;           // [T*U]

  const int WPB = 4;                            // waves per block
  const int PPW = 8;                            // pairs per wave
  int blocks = (npairs + WPB * PPW - 1) / (WPB * PPW);
  tdt_lse_kernel<<<blocks, WPB * WAVE, 0, stream>>>(logits, targets, emit_ws,
                                                    blank_ws, U, D, V, npairs);
  tdt_dp_kernel<<<1, WAVE, 0, stream>>>(emit_ws, blank_ws, tlens,
                                        (float*)d_out, T, U);
}